// GraphSage_49082886258798
// MI455X (gfx1250) — compile-verified
//
#include <hip/hip_runtime.h>
#include <hip/hip_bf16.h>

typedef __attribute__((ext_vector_type(2))) float v2f;
typedef __attribute__((ext_vector_type(8))) float v8f;

#define DH   256   // per-source feature width (both layers)
#define KTOT 512   // concat width = 2*DH
#define KS   32    // K-slab staged through LDS

// ---------------------------------------------------------------------------
// SpMM: neigh[src] += edge_val * feat[dst]   (COO, mean-agg weights precomputed)
// One wave32 per edge; lane l moves 8 floats (2 x float4, at cols 4l and 128+4l).
// Gathers hit the L2-resident feature table; scatters are f32 L2 atomics.
// ---------------------------------------------------------------------------
__global__ void __launch_bounds__(256) spmm_kernel(
    const float* __restrict__ feat,
    const int*   __restrict__ esrc,
    const int*   __restrict__ edst,
    const float* __restrict__ eval,
    float*       __restrict__ out,
    int E)
{
    const int gwave = (int)(((size_t)blockIdx.x * blockDim.x + threadIdx.x) >> 5);
    const int lane  = threadIdx.x & 31;
    if (gwave >= E) return;

    const int   s = esrc[gwave];
    const int   d = edst[gwave];
    const float v = eval[gwave];

    const float4* f = (const float4*)(feat + (size_t)d * DH);
    float*        o = out + (size_t)s * DH;

    const float4 a = f[lane];        // cols [4*lane .. 4*lane+3]
    const float4 b = f[lane + 32];   // cols [128+4*lane .. 128+4*lane+3]
    const int c0 = lane * 4;

    atomicAdd(o + c0 + 0,       a.x * v);
    atomicAdd(o + c0 + 1,       a.y * v);
    atomicAdd(o + c0 + 2,       a.z * v);
    atomicAdd(o + c0 + 3,       a.w * v);
    atomicAdd(o + c0 + 128 + 0, b.x * v);
    atomicAdd(o + c0 + 128 + 1, b.y * v);
    atomicAdd(o + c0 + 128 + 2, b.z * v);
    atomicAdd(o + c0 + 128 + 3, b.w * v);
}

// ---------------------------------------------------------------------------
// Fused: out[r,:] = postproc( [featA | featB][r,:] @ W + bias )
//   postproc = row L2-normalize (/max(||.||,1e-12)), optional ReLU (layer 1).
// One wave per 16-row group, NT = DOUT/16 accumulator tiles = the full row,
// so the norm is finished in-wave via shfl_xor over each 16-lane half.
// fp32 WMMA: v_wmma_f32_16x16x4_f32, K stepped by 4 over KTOT=512.
//   A 16x4 layout : lanes 0-15 hold {K=0,1}, lanes 16-31 hold {K=2,3}, M=lane&15
//   B 4x16 layout : lanes 0-15 hold {K=0,1}, lanes 16-31 hold {K=2,3}, N=lane&15
//   C/D layout    : VGPR i, half h -> row (i + 8h), col = 16*t + (lane&15)
// ---------------------------------------------------------------------------
template<int DOUT, bool RELU>
__global__ void __launch_bounds__(256) sage_gemm_kernel(
    const float* __restrict__ featA,   // [N, DH]  (self features)
    const float* __restrict__ featB,   // [N, DH]  (aggregated neighbors)
    const float* __restrict__ W,       // [KTOT, DOUT] row-major
    const float* __restrict__ bias,    // [DOUT]
    float*       __restrict__ out,     // [N, DOUT]
    int numGroups)                     // N / 16
{
    constexpr int NT = DOUT / 16;
    __shared__ __align__(16) float lds[KS * DOUT];

    const int lane   = threadIdx.x & 31;
    const int lane16 = lane & 15;
    const int half   = lane >> 4;
    const int group  = blockIdx.x * (blockDim.x >> 5) + (threadIdx.x >> 5);
    const bool active = (group < numGroups);          // wave-uniform
    const int  row    = group * 16 + lane16;

    v8f acc[NT];
    const v8f vzero = {};
    #pragma unroll
    for (int t = 0; t < NT; ++t) acc[t] = vzero;

    for (int ks = 0; ks < KTOT; ks += KS) {
        __syncthreads();
        // cooperative stage of the W K-slab into LDS (contiguous, float4)
        {
            const float4* wsrc = (const float4*)(W + (size_t)ks * DOUT);
            float4* ldst = (float4*)lds;
            for (int idx = threadIdx.x; idx < (KS * DOUT) / 4; idx += 256)
                ldst[idx] = wsrc[idx];
        }
        __syncthreads();

        if (active) {
            // slab lies entirely in featA (ks < DH) or featB (ks >= DH)
            const float* fsrc  = (ks < DH) ? featA : featB;
            const int    cbase = (ks < DH) ? ks : (ks - DH);
            const float* arow  = fsrc + (size_t)row * DH + cbase + 2 * half;

            #pragma unroll
            for (int k = 0; k < KS; k += 4) {
                const float2 av = *(const float2*)(arow + k);
                v2f a; a.x = av.x; a.y = av.y;

                const float* bbase = lds + (k + 2 * half) * DOUT + lane16;
                #pragma unroll
                for (int t = 0; t < NT; ++t) {
                    v2f b;
                    b.x = bbase[t * 16];
                    b.y = bbase[t * 16 + DOUT];
                    acc[t] = __builtin_amdgcn_wmma_f32_16x16x4_f32(
                        false, a, false, b, (short)0, acc[t], false, false);
                }
            }
        }
    }

    if (!active) return;

    // bias add + per-row sum of squares (8 rows per lane half)
    float bv[NT];
    #pragma unroll
    for (int t = 0; t < NT; ++t) bv[t] = bias[t * 16 + lane16];

    float p[8];
    #pragma unroll
    for (int i = 0; i < 8; ++i) p[i] = 0.0f;

    #pragma unroll
    for (int t = 0; t < NT; ++t) {
        #pragma unroll
        for (int i = 0; i < 8; ++i) {
            const float c = acc[t][i] + bv[t];
            acc[t][i] = c;
            p[i] += c * c;
        }
    }

    // reduce across the 16 lanes of each half (xor masks < 16 stay in-half)
    #pragma unroll
    for (int m = 1; m < 16; m <<= 1) {
        #pragma unroll
        for (int i = 0; i < 8; ++i)
            p[i] += __shfl_xor(p[i], m, 32);
    }

    float scale[8];
    #pragma unroll
    for (int i = 0; i < 8; ++i)
        scale[i] = 1.0f / fmaxf(sqrtf(p[i]), 1e-12f);

    #pragma unroll
    for (int i = 0; i < 8; ++i) {
        const int m = i + 8 * half;                       // row within group
        float* orow = out + (size_t)(group * 16 + m) * DOUT + lane16;
        #pragma unroll
        for (int t = 0; t < NT; ++t) {
            float v = acc[t][i] * scale[i];
            if (RELU) v = fmaxf(v, 0.0f);
            orow[t * 16] = v;                             // coalesced b32 stores
        }
    }
}

// ---------------------------------------------------------------------------
extern "C" void kernel_launch(void* const* d_in, const int* in_sizes, int n_in,
                              void* d_out, int out_size, void* d_ws, size_t ws_size,
                              hipStream_t stream)
{
    const float* x    = (const float*)d_in[0];
    const int*   esrc = (const int*)  d_in[1];
    const int*   edst = (const int*)  d_in[2];
    const float* eval = (const float*)d_in[3];
    const float* W1   = (const float*)d_in[4];
    const float* b1   = (const float*)d_in[5];
    const float* W2   = (const float*)d_in[6];
    const float* b2   = (const float*)d_in[7];

    const int N = in_sizes[0] / DH;   // 100000
    const int E = in_sizes[1];        // 3200000
    const int groups = N / 16;        // N is a multiple of 16 (6250 groups)

    float* neigh = (float*)d_ws;                  // [N, DH] (reused both layers)
    float* h1    = neigh + (size_t)N * DH;        // [N, DH]
    const size_t featBytes = (size_t)N * DH * sizeof(float);

    const dim3 blk(256);
    const int spmmBlocks = (E + 7) / 8;           // 8 waves/block, 1 edge/wave
    const int gemmBlocks = (groups + 7) / 8;      // 8 row-groups/block

    // ---- layer 1: h1 = relu( normalize( [x | A x] W1 + b1 ) ) ----
    hipMemsetAsync(neigh, 0, featBytes, stream);
    spmm_kernel<<<spmmBlocks, blk, 0, stream>>>(x, esrc, edst, eval, neigh, E);
    sage_gemm_kernel<256, true><<<gemmBlocks, blk, 0, stream>>>(
        x, neigh, W1, b1, h1, groups);

    // ---- layer 2: out = normalize( [h1 | A h1] W2 + b2 ) ----
    hipMemsetAsync(neigh, 0, featBytes, stream);
    spmm_kernel<<<spmmBlocks, blk, 0, stream>>>(h1, esrc, edst, eval, neigh, E);
    sage_gemm_kernel<128, false><<<gemmBlocks, blk, 0, stream>>>(
        h1, neigh, W2, b2, (float*)d_out, groups);
}